// CombinedBandPassFilterSequential_10548439679476
// MI455X (gfx1250) — compile-verified
//
#include <hip/hip_runtime.h>

// ---------------------------------------------------------------------------
// Zero-phase FIR filter bank (filtfilt) for MI455X / gfx1250.
//
// Strategy: compose the two passes into one dense convolution with the
// autocorrelation kernel g = h * reverse(h)  (exact except within K/2 of the
// edges), run that as a bands(16) x time GEMM on V_WMMA_F32_16X16X4_F32, and
// patch the boundary strips with the exact two-pass formula (tiny VALU work).
//
// Compute-bound: ~75 GFLOP fp32 vs ~176 MB of HBM traffic (7.6 us @ 23.3TB/s),
// so everything is organized around feeding the fp32 WMMA pipe from LDS.
// The composed-filter table is staged into LDS once per (persistent) block by
// the Tensor Data Mover (TENSOR_LOAD_TO_LDS + s_wait_tensorcnt).
// ---------------------------------------------------------------------------

typedef __attribute__((ext_vector_type(2))) float v2f;
typedef __attribute__((ext_vector_type(8))) float v8f;
typedef __attribute__((ext_vector_type(4))) unsigned int v4u;
typedef __attribute__((ext_vector_type(8))) int v8i;
typedef __attribute__((ext_vector_type(4))) int v4i;

#define T_LEN 2097152
#define NB    10
#define KP    769
#define KA    129
#define GP    (2*KP - 1)      /* 1537 composed taps (pha) */
#define GA    (2*KA - 1)      /* 257 composed taps (amp)  */
#define GP_PAD 1540           /* padded to multiple of 4  */
#define GA_PAD 260
#define CP    (KP/2)          /* 384 */
#define CA    (KA/2)          /* 64  */

// workspace layout (float offsets)
#define WS_GP    0
#define WS_GA    (WS_GP + 16*GP_PAD)            /* 24640 */
#define WS_Y1LP  (WS_GA + 16*GA_PAD)            /* 28800 */
#define WS_Y1RP  (WS_Y1LP + NB*(2*CP+1))
#define WS_Y1LA  (WS_Y1RP + NB*(2*CP+1))
#define WS_Y1RA  (WS_Y1LA + NB*(2*CA+1))

// ---------------------------------------------------------------------------
// g[b][u] = sum_k h[b][k] * h[b][k - (u-(K-1))]   (autocorrelation, 'full')
// Padded bands (b>=10) and padded taps (u>=G) are zero.
// ---------------------------------------------------------------------------
__global__ void prep_g_kernel(const float* __restrict__ hP,
                              const float* __restrict__ hA,
                              float* __restrict__ ws) {
  int tid = blockIdx.x * blockDim.x + threadIdx.x;
  const int nP = 16 * GP_PAD;
  const int nA = 16 * GA_PAD;
  if (tid < nP) {
    int b = tid / GP_PAD, u = tid % GP_PAD;
    float acc = 0.f;
    if (b < NB) {
      int j = u - (KP - 1);
      const float* h = hP + b * KP;
      for (int k = 0; k < KP; ++k) {
        int km = k - j;
        if (km >= 0 && km < KP) acc += h[k] * h[km];
      }
    }
    ws[WS_GP + tid] = acc;
  } else if (tid < nP + nA) {
    int t2 = tid - nP;
    int b = t2 / GA_PAD, u = t2 % GA_PAD;
    float acc = 0.f;
    if (b < NB) {
      int j = u - (KA - 1);
      const float* h = hA + b * KA;
      for (int k = 0; k < KA; ++k) {
        int km = k - j;
        if (km >= 0 && km < KA) acc += h[k] * h[km];
      }
    }
    ws[WS_GA + t2] = acc;
  }
}

// ---------------------------------------------------------------------------
// Exact pass-1 values y1[b][s] on boundary strips s in [0,2c] and [T-1-2c,T-1]
// (needed for the exact boundary fix of pass 2).
// ---------------------------------------------------------------------------
__device__ __forceinline__ float y1_at(const float* __restrict__ x,
                                       const float* __restrict__ h,
                                       int K, int c, long s) {
  float acc = 0.f;
  for (int k = 0; k < K; ++k) {
    long xi = s + k - c;
    if (xi >= 0 && xi < (long)T_LEN) acc += h[k] * x[xi];
  }
  return acc;
}

__global__ void boundary_y1_kernel(const float* __restrict__ x,
                                   const float* __restrict__ hP,
                                   const float* __restrict__ hA,
                                   float* __restrict__ ws) {
  int tid = blockIdx.x * blockDim.x + threadIdx.x;
  const int LPn = NB * (2*CP + 1);
  const int LAn = NB * (2*CA + 1);
  if (tid < LPn) {                              // left strip, pha
    int b = tid / (2*CP+1), s = tid % (2*CP+1);
    ws[WS_Y1LP + tid] = y1_at(x, hP + b*KP, KP, CP, s);
  } else if (tid < 2*LPn) {                     // right strip, pha
    int e = tid - LPn;
    int b = e / (2*CP+1), i = e % (2*CP+1);
    ws[WS_Y1RP + e] = y1_at(x, hP + b*KP, KP, CP, (long)T_LEN - 1 - 2*CP + i);
  } else if (tid < 2*LPn + LAn) {               // left strip, amp
    int e = tid - 2*LPn;
    int b = e / (2*CA+1), s = e % (2*CA+1);
    ws[WS_Y1LA + e] = y1_at(x, hA + b*KA, KA, CA, s);
  } else if (tid < 2*LPn + 2*LAn) {             // right strip, amp
    int e = tid - 2*LPn - LAn;
    int b = e / (2*CA+1), i = e % (2*CA+1);
    ws[WS_Y1RA + e] = y1_at(x, hA + b*KA, KA, CA, (long)T_LEN - 1 - 2*CA + i);
  }
}

// ---------------------------------------------------------------------------
// Main interior kernel (persistent): out[band][t] = sum_u g[band][u] *
// x[t + u - (K-1)] as a 16(bands) x 16(t) x 4(taps) fp32 WMMA GEMM.
//
// Block = 256 threads = 8 waves; each wave holds 4 accumulators (64 t);
// block covers 512 t per tile and strides over tiles (persistent), so the
// composed-filter table is DMA'd into LDS exactly once per block via the
// Tensor Data Mover.
//
// Fragment addressing (wave32, per CDNA5 ISA 7.12.2):
//   A (16x4 f32): lane m = lane%16, VGPR pair holds K = 2*(lane/16)+{0,1}
//   B (4x16 f32): lane n = lane%16, VGPR v holds row K = v + 2*(lane/16)
//   D (16x16 f32): lane n = lane%16, VGPR v holds row M = v + 8*(lane/16)
// ---------------------------------------------------------------------------
__global__ __launch_bounds__(256)
void conv_wmma_kernel(const float* __restrict__ x,
                      const float* __restrict__ g,
                      float* __restrict__ out,
                      int K, int Gpad, int outBase) {
  extern __shared__ float smem[];
  float* filt = smem;                  // 16 * Gpad floats (filled by TDM)
  float* xwin = smem + 16 * Gpad;      // 512 + Gpad - 1 floats

  const int tid  = threadIdx.x;
  const int W    = 512 + Gpad - 1;
  const int wave = tid >> 5;
  const int lane = tid & 31;
  const int n    = lane & 15;    // column (time) / A row (band)
  const int kh   = lane >> 4;    // lane-half: K offset selector

  // ---- stage composed filters (16 x Gpad fp32, row-contiguous) into LDS ----
#if __has_builtin(__builtin_amdgcn_tensor_load_to_lds) && __has_builtin(__builtin_amdgcn_s_wait_tensorcnt)
  if (wave == 0) {                     // one TDM issue per block (EXEC ignored)
    unsigned long long ga = (unsigned long long)(const void*)g;
    unsigned int ldsOff = __builtin_amdgcn_groupstaticsize();  // dyn-LDS base
    // D# group 0: count=1 | lds_addr | global_addr[56:0] | type=2
    v4u d0;
    d0[0] = 1u;
    d0[1] = ldsOff;
    d0[2] = (unsigned int)(ga & 0xFFFFFFFFu);
    d0[3] = (unsigned int)((ga >> 32) & 0x01FFFFFFu) | (2u << 30);
    // D# group 1: 2D tensor, data_size=4B, dims = Gpad x 16, tile = Gpad x 16
    unsigned int tdim0 = (unsigned int)Gpad;
    v8i d1;
    d1[0] = (int)(2u << 16);                                   // data_size=4B
    d1[1] = (int)((tdim0 & 0xFFFFu) << 16);                    // tdim0[15:0]
    d1[2] = (int)((tdim0 >> 16) | (16u << 16));                // tdim0 hi | tdim1=16
    d1[3] = (int)(tdim0 << 16);                                // tile_dim0 = Gpad
    d1[4] = 16;                                                // tile_dim1 = 16
    d1[5] = (int)tdim0;                                        // dim0_stride = Gpad
    d1[6] = 0;
    d1[7] = 0;
    v4i dz4 = {0, 0, 0, 0};
    v8i dz8 = {0, 0, 0, 0, 0, 0, 0, 0};
    __builtin_amdgcn_tensor_load_to_lds(d0, d1, dz4, dz4, dz8, 0);
    __builtin_amdgcn_s_wait_tensorcnt(0);
  }
#else
  for (int i = tid; i < 16 * Gpad; i += 256) filt[i] = g[i];
#endif
  __syncthreads();

  const float* arow = filt + n * Gpad + 2 * kh;      // 8B-aligned (Gpad%4==0)
  const float* brow = xwin + wave * 64 + n + 2 * kh;

  const int NT = T_LEN / 512;
  for (int tile = blockIdx.x; tile < NT; tile += gridDim.x) {
    const long tbase = (long)tile * 512;
    const long t0    = tbase - (K - 1);              // x index of xwin[0]
    for (int i = tid; i < W; i += 256) {
      long gi = t0 + i;
      xwin[i] = (gi >= 0 && gi < (long)T_LEN) ? x[gi] : 0.f;
    }
    __syncthreads();

    v8f acc0 = {0,0,0,0,0,0,0,0};
    v8f acc1 = {0,0,0,0,0,0,0,0};
    v8f acc2 = {0,0,0,0,0,0,0,0};
    v8f acc3 = {0,0,0,0,0,0,0,0};

    for (int kc = 0; kc < Gpad; kc += 4) {
      v2f a = *(const v2f*)(arow + kc);              // A: g[m][kc+2kh .. +1]
      const float* bp = brow + kc;
      v2f b0, b1, b2, b3;                            // B: x window, 4 t-tiles
      b0.x = bp[0];  b0.y = bp[1];
      b1.x = bp[16]; b1.y = bp[17];
      b2.x = bp[32]; b2.y = bp[33];
      b3.x = bp[48]; b3.y = bp[49];
      acc0 = __builtin_amdgcn_wmma_f32_16x16x4_f32(false, a, false, b0, (short)0, acc0, false, false);
      acc1 = __builtin_amdgcn_wmma_f32_16x16x4_f32(false, a, false, b1, (short)0, acc1, false, false);
      acc2 = __builtin_amdgcn_wmma_f32_16x16x4_f32(false, a, false, b2, (short)0, acc2, false, false);
      acc3 = __builtin_amdgcn_wmma_f32_16x16x4_f32(false, a, false, b3, (short)0, acc3, false, false);
    }

    const long tb = tbase + wave * 64 + n;
    v8f accs[4] = {acc0, acc1, acc2, acc3};
#pragma unroll
    for (int j = 0; j < 4; ++j) {
#pragma unroll
      for (int v = 0; v < 8; ++v) {
        int m = v + 8 * kh;                          // D-matrix row = band
        if (m < NB) out[(long)(outBase + m) * T_LEN + tb + j * 16] = accs[j][v];
      }
    }
    __syncthreads();   // protect xwin before the next tile overwrites it
  }
}

// ---------------------------------------------------------------------------
// Exact boundary fix: y2[b][t] = sum_k h[b][k] * y1clip[b][t + c - k]
// for t in [0,c) and [T-c, T), using the stored boundary y1 strips.
// ---------------------------------------------------------------------------
__global__ void boundary_fix_kernel(const float* __restrict__ hP,
                                    const float* __restrict__ hA,
                                    const float* __restrict__ ws,
                                    float* __restrict__ out) {
  int tid = blockIdx.x * blockDim.x + threadIdx.x;
  const int nLP = NB * CP, nLA = NB * CA;
  int bank, b, t, K, c, y1off, y1stride, right;
  if (tid < nLP)                { bank=0; right=0; b=tid/CP;        t=tid%CP;        K=KP; c=CP; y1off=WS_Y1LP; }
  else if (tid < 2*nLP)         { bank=0; right=1; b=(tid-nLP)/CP;  t=(tid-nLP)%CP;  K=KP; c=CP; y1off=WS_Y1RP; }
  else if (tid < 2*nLP+nLA)     { bank=1; right=0; b=(tid-2*nLP)/CA; t=(tid-2*nLP)%CA; K=KA; c=CA; y1off=WS_Y1LA; }
  else if (tid < 2*nLP+2*nLA)   { bank=1; right=1; b=(tid-2*nLP-nLA)/CA; t=(tid-2*nLP-nLA)%CA; K=KA; c=CA; y1off=WS_Y1RA; }
  else return;
  y1stride = 2*c + 1;
  const float* h = (bank == 0) ? (hP + b*KP) : (hA + b*KA);
  long tt = right ? ((long)T_LEN - 1 - t) : (long)t;   // actual output time
  float acc = 0.f;
  for (int k = 0; k < K; ++k) {
    long s = tt + c - k;                               // y1 sample index
    if (s < 0 || s >= (long)T_LEN) continue;
    long i = right ? (s - ((long)T_LEN - 1 - 2*c)) : s; // index into strip
    acc += h[k] * ws[y1off + b * y1stride + (int)i];
  }
  out[(long)(bank * NB + b) * T_LEN + tt] = acc;
}

// ---------------------------------------------------------------------------
extern "C" void kernel_launch(void* const* d_in, const int* in_sizes, int n_in,
                              void* d_out, int out_size, void* d_ws, size_t ws_size,
                              hipStream_t stream) {
  const float* x  = (const float*)d_in[0];   // (1,1,T)
  const float* hP = (const float*)d_in[1];   // (10,769)
  const float* hA = (const float*)d_in[2];   // (10,129)
  float* out = (float*)d_out;                // 20*T
  float* ws  = (float*)d_ws;

  // 1) composed autocorrelation filters (padded 16 bands x Gpad taps)
  {
    int n = 16*GP_PAD + 16*GA_PAD;
    prep_g_kernel<<<(n + 255)/256, 256, 0, stream>>>(hP, hA, ws);
  }
  // 2) exact pass-1 boundary strips
  {
    int n = 2 * (NB*(2*CP+1) + NB*(2*CA+1));
    boundary_y1_kernel<<<(n + 255)/256, 256, 0, stream>>>(x, hP, hA, ws);
  }
  // 3) dense WMMA convolution for both banks (persistent, writes every sample)
  {
    const int grid = 1024;             // persistent blocks; tiles = 4096
    size_t smemP = (size_t)(16*GP_PAD + 512 + GP_PAD - 1) * sizeof(float);
    size_t smemA = (size_t)(16*GA_PAD + 512 + GA_PAD - 1) * sizeof(float);
    conv_wmma_kernel<<<grid, 256, smemP, stream>>>(x, ws + WS_GP, out, KP, GP_PAD, 0);
    conv_wmma_kernel<<<grid, 256, smemA, stream>>>(x, ws + WS_GA, out, KA, GA_PAD, NB);
  }
  // 4) overwrite the c-wide edge strips with the exact two-pass values
  {
    int n = 2 * (NB*CP + NB*CA);
    boundary_fix_kernel<<<(n + 255)/256, 256, 0, stream>>>(hP, hA, ws, out);
  }
}